// CtcLoss_70068096467498
// MI455X (gfx1250) — compile-verified
//
#include <hip/hip_runtime.h>
#include <hip/hip_bf16.h>

#ifndef __has_builtin
#define __has_builtin(x) 0
#endif

#if __has_builtin(__builtin_amdgcn_global_load_async_to_lds_b128) && \
    __has_builtin(__builtin_amdgcn_s_wait_asynccnt)
#define USE_ASYNC_LDS 1
#else
#define USE_ASYNC_LDS 0
#endif

typedef float v2f __attribute__((ext_vector_type(2)));
typedef float v8f __attribute__((ext_vector_type(8)));
typedef int   v4i __attribute__((__vector_size__(4 * sizeof(int))));

#define NEGV (-1e30f)

// Problem constants (fixed by the reference setup)
#define CB 32
#define CT 2048
#define CV 128
#define CS 256
#define CL (2 * CS + 1)   // 513

// ---------------------------------------------------------------------------
// Kernel 1: per-(b,t) log-softmax normalizer norm[row] = log(sum_v exp(x[v]))
// Row sums computed on the matrix pipe: D = exp(A) x Ones via
// v_wmma_f32_16x16x4_f32, 16 rows per wave, K swept 128 in chunks of 4.
// ---------------------------------------------------------------------------
__global__ __launch_bounds__(256) void ctc_norm_wmma(const float* __restrict__ pred,
                                                     float* __restrict__ norm) {
  const int lane = threadIdx.x & 31;
  const int wid  = threadIdx.x >> 5;
  const int wave = blockIdx.x * 8 + wid;     // 8 waves/block
  const int base_row = wave * 16;            // 16 rows per wave, rows = B*T
  // 32-bit A 16x4 layout: lanes 0-15 hold (M=lane, K=0..1 in v0..v1),
  // lanes 16-31 hold (M=lane-16, K=2..3).
  const int mrow = base_row + (lane & 15);
  const int koff = (lane & 16) ? 2 : 0;
  const float* p = pred + (size_t)mrow * CV + koff;

  v2f ones; ones.x = 1.0f; ones.y = 1.0f;   // B = 4x16 all-ones (layout-invariant)
  v8f c = {};                                // f32 16x16 accumulator, 8 VGPRs

#pragma unroll
  for (int k0 = 0; k0 < CV; k0 += 4) {
    v2f a;
    a.x = __expf(p[k0]);
    a.y = __expf(p[k0 + 1]);
    c = __builtin_amdgcn_wmma_f32_16x16x4_f32(
        /*neg_a=*/false, a, /*neg_b=*/false, ones,
        /*c_mod=*/(short)0, c, /*reuse_a=*/false, /*reuse_b=*/false);
  }

  // D layout: VGPR r, lanes 0-15 -> M=r,N=lane ; lanes 16-31 -> M=8+r,N=lane-16.
  // Lane 0 column N=0 holds rowsum[0..7]; lane 16 holds rowsum[8..15].
  if ((lane & 15) == 0) {
    const int rbase = base_row + ((lane >> 4) ? 8 : 0);
#pragma unroll
    for (int r = 0; r < 8; ++r) norm[rbase + r] = __logf(c[r]);
  }
}

// ---------------------------------------------------------------------------
// Async stage of one 128-float log-prob row into LDS (wave 0 only: 32 lanes
// x b128 = 512 bytes).
// ---------------------------------------------------------------------------
__device__ __forceinline__ void stage_row(const float* gsrc, float* ldst, int lane) {
#if USE_ASYNC_LDS
  __builtin_amdgcn_global_load_async_to_lds_b128(
      (v4i __attribute__((address_space(1)))*)(gsrc + lane * 4),
      (v4i __attribute__((address_space(3)))*)(ldst + lane * 4),
      /*offset=*/0, /*cpol=*/0);
#else
  const float4 v = *(const float4*)(gsrc + lane * 4);
  *(float4*)(ldst + lane * 4) = v;
#endif
}

__device__ __forceinline__ void wait_stage() {
#if USE_ASYNC_LDS
  __builtin_amdgcn_s_wait_asynccnt(0);
#endif
}

// ---------------------------------------------------------------------------
// Kernel 2: CTC alpha recursion. One block per batch element; thread s owns
// extended-lattice cell s (L=513). Alpha double-buffered in LDS; log-prob row
// double-buffered in LDS with async global->LDS prefetch one step ahead.
// ---------------------------------------------------------------------------
__global__ __launch_bounds__(544) void ctc_dp(const float* __restrict__ pred,
                                              const int* __restrict__ plen,
                                              const int* __restrict__ gt,
                                              const int* __restrict__ glen,
                                              const float* __restrict__ norm,
                                              float* __restrict__ partial) {
  // lprow first + 16B alignment: LDS b128 async writes require 16-byte
  // natural alignment (ISA alignment rules).
  __shared__ __attribute__((aligned(16))) float lprow[2][CV];
  __shared__ __attribute__((aligned(16))) float alpha[2][CL];

  const int b   = blockIdx.x;
  const int tid = threadIdx.x;
  const float* prow = pred + (size_t)b * CT * CV;
  const float* nrow = norm + (size_t)b * CT;
  const int Tb = plen[b];

  // Extended label + skip mask for this cell (registers only).
  int  exts   = 0;       // blank
  bool skip_s = false;
  if (tid < CL && (tid & 1)) {
    const int j = tid >> 1;
    exts = gt[b * CS + j];
    if (tid >= 2) skip_s = (exts != gt[b * CS + j - 1]);
  }

  // Stage row 0, init alpha.
  if (tid < 32) { stage_row(prow, lprow[0], tid); wait_stage(); }
  __syncthreads();
  const float n0 = nrow[0];
  if (tid < CL) {
    float a = NEGV;
    if (tid < 2) a = lprow[0][exts] - n0;   // s=0 blank, s=1 first label
    alpha[0][tid] = a;
  }
  if (Tb > 1 && tid < 32) stage_row(prow + CV, lprow[1], tid);  // prefetch row 1
  __syncthreads();

  for (int t = 1; t < Tb; ++t) {
    const int cur = t & 1, prv = cur ^ 1;
    if (tid < 32) wait_stage();             // row t landed in lprow[cur]
    __syncthreads();
    const float normt = nrow[t];
    if (tid < CL) {
      const float a1 = alpha[prv][tid];
      const float a2 = (tid >= 1) ? alpha[prv][tid - 1] : NEGV;
      const float a3 = skip_s ? alpha[prv][tid - 2] : NEGV;
      const float m  = fmaxf(a1, fmaxf(a2, a3));
      const float ls = __expf(a1 - m) + __expf(a2 - m) + __expf(a3 - m);
      alpha[cur][tid] = m + __logf(ls) + (lprow[cur][exts] - normt);
    }
    if (t + 1 < Tb && tid < 32)             // prefetch row t+1 into other buffer
      stage_row(prow + (size_t)(t + 1) * CV, lprow[prv], tid);
    __syncthreads();
  }

  if (tid == 0) {
    const int   fin = (Tb - 1) & 1;
    const int   U   = glen[b];
    const float aL  = alpha[fin][2 * U];
    const float aM  = alpha[fin][2 * U - 1];
    const float m   = fmaxf(aL, aM);
    float loss = -(m + __logf(__expf(aL - m) + __expf(aM - m)));
    if (!(loss < 1e10f)) loss = 0.0f;       // zero_infinity (also kills NaN)
    partial[b] = loss / (float)U;
  }
}

// ---------------------------------------------------------------------------
// Kernel 3: deterministic batch-mean reduction (single wave32).
// ---------------------------------------------------------------------------
__global__ __launch_bounds__(32) void ctc_reduce(const float* __restrict__ partial,
                                                 float* __restrict__ out) {
  float v = (threadIdx.x < CB) ? partial[threadIdx.x] : 0.0f;
#pragma unroll
  for (int off = 16; off > 0; off >>= 1) v += __shfl_down(v, off, 32);
  if (threadIdx.x == 0) out[0] = v / (float)CB;
}

extern "C" void kernel_launch(void* const* d_in, const int* in_sizes, int n_in,
                              void* d_out, int out_size, void* d_ws, size_t ws_size,
                              hipStream_t stream) {
  const float* pred = (const float*)d_in[0];   // [B,T,V] f32
  const int*   plen = (const int*)d_in[1];     // [B]
  const int*   gt   = (const int*)d_in[2];     // [B,S]
  const int*   glen = (const int*)d_in[3];     // [B]
  float* out = (float*)d_out;

  float* norm    = (float*)d_ws;               // B*T floats (256 KB)
  float* partial = norm + (size_t)CB * CT;     // B floats

  // Phase 1: 65536 rows, 16 rows/wave, 8 waves/block -> 512 blocks.
  ctc_norm_wmma<<<512, 256, 0, stream>>>(pred, norm);
  // Phase 2: one workgroup per batch element, 544 threads (>= L=513).
  ctc_dp<<<CB, 544, 0, stream>>>(pred, plen, gt, glen, norm, partial);
  // Phase 3: batch mean.
  ctc_reduce<<<1, 32, 0, stream>>>(partial, out);
}